// ClassAttention_13941463842886
// MI455X (gfx1250) — compile-verified
//
#include <hip/hip_runtime.h>

// ---------------------------------------------------------------------------
// ClassAttention (single CLS query) restructured to avoid the full kv GEMM:
//   logit[b,h,n] = r[b,h] . x[b,n] + c0[b,h],  r = SCALE * q^T Wk   (pass 1)
//   s[b,h,:]     = sum_n softmax(logit)[n] * x[b,n,:]               (pass 2)
//   cls[b,h]     = s[b,h] . Wv[h]^T + bv ;  out = cls @ proj_w^T + proj_b
// ~4 GFLOP instead of 237 GFLOP; memory-bound at ~2 passes over x
// (616 MB -> ~26us @ 23.3 TB/s). Big passes: fp32 WMMA (V_WMMA_F32_16X16X4_F32)
// fed by double-buffered GLOBAL_LOAD_ASYNC_TO_LDS_B128 (ASYNCcnt pipeline).
// ---------------------------------------------------------------------------

#define BATCH 32
#define NTOK  3137
#define CDIM  768
#define HEADS 12
#define HDIM  64
#define SCALEF 0.125f   // 64^-0.5

typedef __attribute__((ext_vector_type(2))) float v2f;
typedef __attribute__((ext_vector_type(8))) float v8f;

#define KC   128   // K-chunk (fp32 elems) staged in LDS for both GEMMs
#define LSTR 132   // LDS row stride (floats): 128 + 4 pad -> conflict-free
#define XSTR 20    // LDS stride for [token][col] tile in pass 2

// low 32 bits of a flat LDS pointer == LDS byte offset (LDS aperture rule)
__device__ __forceinline__ unsigned lds_addr32(const void* p) {
  return (unsigned)(unsigned long long)p;
}

// async DMA: 32 lanes x 16B -> LDS, tracked by ASYNCcnt (no VGPR round-trip)
__device__ __forceinline__ void async_load_b128(unsigned lds_off, const void* gaddr) {
  asm volatile("global_load_async_to_lds_b128 %0, %1, off"
               :: "v"(lds_off), "v"(gaddr) : "memory");
}

#define S_WAIT_ASYNCCNT(n) asm volatile("s_wait_asynccnt " n ::: "memory")
#define S_WAIT_DSCNT0()    asm volatile("s_wait_dscnt 0x0" ::: "memory")

// ---------------------------------------------------------------------------
// Pass 0: q[b,h,:] = x[b,0,:] @ q_w^T + q_b ;  r = SCALE * q^T Wk ; c0 = SCALE q.bk
// ---------------------------------------------------------------------------
__global__ void ca_prep_kernel(const float* __restrict__ x,
                               const float* __restrict__ kv_w,
                               const float* __restrict__ kv_b,
                               const float* __restrict__ q_w,
                               const float* __restrict__ q_b,
                               float* __restrict__ r,    // [B,H,C]
                               float* __restrict__ c0) { // [B,H]
  int bh = blockIdx.x;
  int b = bh / HEADS, h = bh % HEADS;
  int tid = threadIdx.x;                 // 64 threads = HDIM
  __shared__ float qs[HDIM];

  const float* xrow = x + (size_t)b * NTOK * CDIM;   // token 0 of batch b
  int qrow = h * HDIM + tid;
  float acc = 0.f;
  for (int c = 0; c < CDIM; ++c) acc += xrow[c] * q_w[(size_t)qrow * CDIM + c];
  qs[tid] = acc + q_b[qrow];
  __syncthreads();

  if (tid == 0) {
    float s = 0.f;
    for (int d = 0; d < HDIM; ++d) s += qs[d] * kv_b[h * HDIM + d];
    c0[bh] = SCALEF * s;
  }
  for (int c = tid; c < CDIM; c += HDIM) {
    float s = 0.f;
    for (int d = 0; d < HDIM; ++d)
      s += qs[d] * kv_w[(size_t)(h * HDIM + d) * CDIM + c];
    r[(size_t)bh * CDIM + c] = SCALEF * s;
  }
}

// ---------------------------------------------------------------------------
// Pass 1: logits tile = R[b](16x768) x X[b]^T(768x16) via f32 WMMA.
// One wave per (batch, 16-token tile). Double-buffered async global->LDS.
// Out-of-range heads/tokens are CLAMPED, not zeroed: D row m depends only on
// A row m, D col n only on B col n, and we never store m>=12 / n>=N.
// ---------------------------------------------------------------------------
__global__ void ca_logits_kernel(const float* __restrict__ x,
                                 const float* __restrict__ r,
                                 const float* __restrict__ c0,
                                 float* __restrict__ logits) { // [B,H,N]
  int n0   = blockIdx.x * 16;
  int b    = blockIdx.y;
  int lane = threadIdx.x;        // 32
  int row  = lane & 15;
  int hi   = lane >> 4;          // 0 or 1

  __shared__ float Rs[2][16 * LSTR];
  __shared__ float Xs[2][16 * LSTR];

  const float* xb = x + (size_t)b * NTOK * CDIM;
  const float* rb = r + (size_t)b * HEADS * CDIM;

  const int lcol = lane * 4;     // per-lane float offset within a 128-float row
  unsigned rbase[2] = { lds_addr32(&Rs[0][0]), lds_addr32(&Rs[1][0]) };
  unsigned xbase[2] = { lds_addr32(&Xs[0][0]), lds_addr32(&Xs[1][0]) };

  auto issue = [&](int c, int buf) {
    int k0 = c * KC;
#pragma unroll
    for (int i = 0; i < 16; ++i) {
      int rrow = (i < HEADS) ? i : (HEADS - 1);          // clamp pad heads
      async_load_b128(rbase[buf] + (unsigned)(i * LSTR + lcol) * 4,
                      rb + (size_t)rrow * CDIM + k0 + lcol);
      int n = n0 + i; n = (n < NTOK) ? n : (NTOK - 1);   // clamp pad tokens
      async_load_b128(xbase[buf] + (unsigned)(i * LSTR + lcol) * 4,
                      xb + (size_t)n * CDIM + k0 + lcol);
    }
  };

  v8f acc = {};
  const int NCHUNK = CDIM / KC;  // 6
  issue(0, 0);
  for (int c = 0; c < NCHUNK; ++c) {
    int buf = c & 1;
    if (c + 1 < NCHUNK) {
      S_WAIT_DSCNT0();                 // prior reads of target buffer finished
      issue(c + 1, buf ^ 1);
      S_WAIT_ASYNCCNT("0x20");         // in-order: first 32 (= chunk c) landed
    } else {
      S_WAIT_ASYNCCNT("0x0");
    }
    const float* RsB = &Rs[buf][0];
    const float* XsB = &Xs[buf][0];
    // A 16x4 f32: lanes 0-15 M=lane K={k,k+1}; lanes 16-31 K={k+2,k+3}
    // B 4x16 f32: lanes 0-15 N=lane K={k,k+1}; lanes 16-31 K={k+2,k+3}
#pragma unroll
    for (int kk = 0; kk < KC; kk += 4) {
      v2f a  = *(const v2f*)(&RsB[row * LSTR + kk + 2 * hi]);
      v2f bb = *(const v2f*)(&XsB[row * LSTR + kk + 2 * hi]);
      acc = __builtin_amdgcn_wmma_f32_16x16x4_f32(false, a, false, bb,
                                                  (short)0, acc, false, false);
    }
  }

  // D layout: vgpr rr, lanes 0-15 -> (M=rr, N=lane); lanes 16-31 -> (M=rr+8)
  int n = n0 + row;
  if (n < NTOK) {
#pragma unroll
    for (int rr = 0; rr < 8; ++rr) {
      int m = rr + 8 * hi;
      if (m < HEADS)
        logits[((size_t)b * HEADS + m) * NTOK + n] = acc[rr] + c0[b * HEADS + m];
    }
  }
}

// ---------------------------------------------------------------------------
// Pass 1.5: in-place softmax over N per (b,h)
// ---------------------------------------------------------------------------
__global__ void ca_softmax_kernel(float* __restrict__ attn) { // [B*H, N]
  float* rowp = attn + (size_t)blockIdx.x * NTOK;
  int tid = threadIdx.x;                  // 256
  __shared__ float red[256];

  float m = -3.402823466e38f;
  for (int n = tid; n < NTOK; n += 256) m = fmaxf(m, rowp[n]);
  red[tid] = m; __syncthreads();
  for (int s = 128; s > 0; s >>= 1) {
    if (tid < s) red[tid] = fmaxf(red[tid], red[tid + s]);
    __syncthreads();
  }
  m = red[0];
  __syncthreads();

  float sum = 0.f;
  for (int n = tid; n < NTOK; n += 256) {
    float e = __expf(rowp[n] - m);
    rowp[n] = e;
    sum += e;
  }
  red[tid] = sum; __syncthreads();
  for (int s = 128; s > 0; s >>= 1) {
    if (tid < s) red[tid] += red[tid + s];
    __syncthreads();
  }
  float inv = 1.f / red[0];
  for (int n = tid; n < NTOK; n += 256) rowp[n] *= inv;
}

// ---------------------------------------------------------------------------
// Pass 2: S[b] (12x768) = Attn[b] (12x3137) x X[b] (3137x768) via f32 WMMA.
// One wave per (batch, 16-column tile); K = tokens, chunked by 128.
// X chunks via double-buffered async DMA (clamped rows: masked by A zeros).
// Attn chunk staged synchronously (small, L2-resident) with true zero-fill
// for past-end tokens (those K-columns DO mix into valid outputs).
// ---------------------------------------------------------------------------
__global__ void ca_wsum_kernel(const float* __restrict__ x,
                               const float* __restrict__ attn, // [B*H, N]
                               float* __restrict__ S) {        // [B,H,C]
  int c0col = blockIdx.x * 16;
  int b     = blockIdx.y;
  int lane  = threadIdx.x;
  int row   = lane & 15;
  int hi    = lane >> 4;

  __shared__ float As[16 * LSTR];   // [head][token-in-chunk]
  __shared__ float Xs[2][KC * XSTR];// [token-in-chunk][col], double buffered

  const float* xb = x + (size_t)b * NTOK * CDIM;
  const float* ab = attn + (size_t)b * HEADS * NTOK;
  unsigned xbase[2] = { lds_addr32(&Xs[0][0]), lds_addr32(&Xs[1][0]) };

  const int subrow = lane >> 2;        // 0..7  (token row within group of 8)
  const int ccol   = (lane & 3) * 4;   // 0,4,8,12 (float col)

  auto issueX = [&](int c, int buf) {
    int nbase = c * KC;
#pragma unroll
    for (int i = 0; i < KC; i += 8) {
      int rr = i + subrow;
      int n = nbase + rr; n = (n < NTOK) ? n : (NTOK - 1);  // clamp; A is 0 there
      async_load_b128(xbase[buf] + (unsigned)(rr * XSTR + ccol) * 4,
                      xb + (size_t)n * CDIM + c0col + ccol);
    }
  };

  v8f acc = {};
  const int NCH = (NTOK + KC - 1) / KC;   // 25
  issueX(0, 0);
  for (int c = 0; c < NCH; ++c) {
    int buf = c & 1;
    int nbase = c * KC;
    // stage attention chunk rows 0..11; rows 12..15 never affect stored output
#pragma unroll
    for (int i = 0; i < HEADS; ++i) {
#pragma unroll
      for (int cc = 0; cc < KC; cc += 32) {
        int n = nbase + cc + lane;
        int nc = (n < NTOK) ? n : (NTOK - 1);
        float v = ab[(size_t)i * NTOK + nc];
        As[i * LSTR + cc + lane] = (n < NTOK) ? v : 0.f;
      }
    }
    if (c + 1 < NCH) {
      S_WAIT_DSCNT0();
      issueX(c + 1, buf ^ 1);
      S_WAIT_ASYNCCNT("0x10");           // chunk c (16 ops) landed
    } else {
      S_WAIT_ASYNCCNT("0x0");
    }
    const float* XsB = &Xs[buf][0];
#pragma unroll
    for (int kk = 0; kk < KC; kk += 4) {
      v2f a = *(const v2f*)(&As[row * LSTR + kk + 2 * hi]);
      v2f bb;
      bb.x = XsB[(kk + 2 * hi)     * XSTR + row];   // B[k][n]: n across lanes
      bb.y = XsB[(kk + 2 * hi + 1) * XSTR + row];
      acc = __builtin_amdgcn_wmma_f32_16x16x4_f32(false, a, false, bb,
                                                  (short)0, acc, false, false);
    }
  }

#pragma unroll
  for (int rr = 0; rr < 8; ++rr) {
    int m = rr + 8 * hi;
    if (m < HEADS)
      S[((size_t)b * HEADS + m) * CDIM + c0col + row] = acc[rr];
  }
}

// ---------------------------------------------------------------------------
// Pass 3: cls[b, h*64+d] = S[b,h,:] . Wv_row + bv   (v half of kv_w)
// ---------------------------------------------------------------------------
__global__ void ca_cls_kernel(const float* __restrict__ S,
                              const float* __restrict__ kv_w,
                              const float* __restrict__ kv_b,
                              float* __restrict__ cls) { // [B, C]
  int bh = blockIdx.x;
  int b = bh / HEADS, h = bh % HEADS;
  int d = threadIdx.x;                    // 64
  const float* srow = S + (size_t)bh * CDIM;
  int wrow = CDIM + h * HDIM + d;         // v rows live at [C, 2C)
  float acc = 0.f;
  for (int c = 0; c < CDIM; ++c) acc += srow[c] * kv_w[(size_t)wrow * CDIM + c];
  cls[(size_t)b * CDIM + h * HDIM + d] = acc + kv_b[wrow];
}

// ---------------------------------------------------------------------------
// Pass 4: out[b,:] = cls[b,:] @ proj_w^T + proj_b
// ---------------------------------------------------------------------------
__global__ void ca_proj_kernel(const float* __restrict__ cls,
                               const float* __restrict__ proj_w,
                               const float* __restrict__ proj_b,
                               float* __restrict__ out) { // [B, C]
  int b = blockIdx.x;
  __shared__ float cs[CDIM];
  for (int c = threadIdx.x; c < CDIM; c += blockDim.x)
    cs[c] = cls[(size_t)b * CDIM + c];
  __syncthreads();
  for (int j = threadIdx.x; j < CDIM; j += blockDim.x) {
    float acc = 0.f;
    for (int c = 0; c < CDIM; ++c) acc += cs[c] * proj_w[(size_t)j * CDIM + c];
    out[(size_t)b * CDIM + j] = acc + proj_b[j];
  }
}

// ---------------------------------------------------------------------------
extern "C" void kernel_launch(void* const* d_in, const int* in_sizes, int n_in,
                              void* d_out, int out_size, void* d_ws, size_t ws_size,
                              hipStream_t stream) {
  const float* x      = (const float*)d_in[0];
  const float* kv_w   = (const float*)d_in[1];
  const float* kv_b   = (const float*)d_in[2];
  const float* q_w    = (const float*)d_in[3];
  const float* q_b    = (const float*)d_in[4];
  const float* proj_w = (const float*)d_in[5];
  const float* proj_b = (const float*)d_in[6];
  float* out = (float*)d_out;

  // workspace layout (floats): total ~1.82M floats (~7.3 MB)
  float* ws   = (float*)d_ws;
  float* r    = ws;                                      // B*H*C
  float* c0   = r    + (size_t)BATCH * HEADS * CDIM;     // B*H
  float* attn = c0   + (size_t)BATCH * HEADS;            // B*H*N
  float* S    = attn + (size_t)BATCH * HEADS * NTOK;     // B*H*C
  float* cls  = S    + (size_t)BATCH * HEADS * CDIM;     // B*C

  ca_prep_kernel<<<dim3(BATCH * HEADS), dim3(HDIM), 0, stream>>>(
      x, kv_w, kv_b, q_w, q_b, r, c0);

  dim3 gLog((NTOK + 15) / 16, BATCH);
  ca_logits_kernel<<<gLog, dim3(32), 0, stream>>>(x, r, c0, attn);

  ca_softmax_kernel<<<dim3(BATCH * HEADS), dim3(256), 0, stream>>>(attn);

  dim3 gWs(CDIM / 16, BATCH);
  ca_wsum_kernel<<<gWs, dim3(32), 0, stream>>>(x, attn, S);

  ca_cls_kernel<<<dim3(BATCH * HEADS), dim3(HDIM), 0, stream>>>(S, kv_w, kv_b, cls);

  ca_proj_kernel<<<dim3(BATCH), dim3(256), 0, stream>>>(cls, proj_w, proj_b, out);
}